// NodeEdgeRefDomain_50869592655561
// MI455X (gfx1250) — compile-verified
//
#include <hip/hip_runtime.h>

// GINE layer for MI455X (gfx1250, wave32):
//   msg_u = relu(x[v]+ea), msg_v = relu(x[u]+ea); scatter-add; then
//   out = relu(((1+eps)x + nbr) @ W1 + b1) @ W2 + b2
// GEMMs use V_WMMA_F32_16X16X4_F32 (wave32 WMMA).

#define DIM 128
#define HIDN 128
#define SPAD 132  // padded LDS row stride (floats) to avoid 64-bank conflicts

typedef __attribute__((ext_vector_type(2))) float v2f;
typedef __attribute__((ext_vector_type(8))) float v8f;

// ---------------------------------------------------------------- zero acc
__global__ void gine_zero_kernel(float4* __restrict__ acc, int n4) {
  int i = blockIdx.x * blockDim.x + threadIdx.x;
  if (i < n4) acc[i] = make_float4(0.f, 0.f, 0.f, 0.f);
}

// ------------------------------------------------------- edge scatter-add
// One wave (32 lanes) per edge; each lane owns 4 consecutive channels.
__global__ void gine_scatter_kernel(const float* __restrict__ x,
                                    const float* __restrict__ ea,
                                    const int* __restrict__ ei,
                                    float* __restrict__ acc, int E) {
  long gid = (long)blockIdx.x * blockDim.x + threadIdx.x;
  long e = gid >> 5;
  if (e >= E) return;
  int lane = (int)(gid & 31);
  int c = lane * 4;

  int u = ei[e];
  int v = ei[e + E];

  float4 a  = *(const float4*)(ea + e * (long)DIM + c);
  float4 xv = *(const float4*)(x + (long)v * DIM + c);
  float4 xu = *(const float4*)(x + (long)u * DIM + c);

  float* pu = acc + (long)u * DIM + c;
  atomicAdd(pu + 0, fmaxf(xv.x + a.x, 0.f));
  atomicAdd(pu + 1, fmaxf(xv.y + a.y, 0.f));
  atomicAdd(pu + 2, fmaxf(xv.z + a.z, 0.f));
  atomicAdd(pu + 3, fmaxf(xv.w + a.w, 0.f));

  float* pv = acc + (long)v * DIM + c;
  atomicAdd(pv + 0, fmaxf(xu.x + a.x, 0.f));
  atomicAdd(pv + 1, fmaxf(xu.y + a.y, 0.f));
  atomicAdd(pv + 2, fmaxf(xu.z + a.z, 0.f));
  atomicAdd(pv + 3, fmaxf(xu.w + a.w, 0.f));
}

// --------------------------------------------------- fused node MLP (WMMA)
// 256 threads = 8 waves; wave w owns output columns [16w, 16w+16).
// Persistent workgroups stride over 16-row node blocks.
__global__ __launch_bounds__(256)
void gine_mlp_kernel(const float* __restrict__ x, const float* __restrict__ acc,
                     const float* __restrict__ W1, const float* __restrict__ b1,
                     const float* __restrict__ W2, const float* __restrict__ b2,
                     const float* __restrict__ eps_p, float* __restrict__ out,
                     int nblocks) {
  __shared__ float sW1[DIM * HIDN];   // 64 KB
  __shared__ float sW2[HIDN * DIM];   // 64 KB
  __shared__ float sIn[16 * SPAD];    // input tile (padded)
  __shared__ float sHid[16 * SPAD];   // hidden tile (padded)

  const int tid  = threadIdx.x;
  const int wave = tid >> 5;
  const int lane = tid & 31;

  // Stage weights into LDS (vectorized, L2-hot).
  {
    const float4* w1v = (const float4*)W1;
    const float4* w2v = (const float4*)W2;
    float4* s1v = (float4*)sW1;
    float4* s2v = (float4*)sW2;
    for (int i = tid; i < DIM * HIDN / 4; i += 256) {
      s1v[i] = w1v[i];
      s2v[i] = w2v[i];
    }
  }

  const float eps1  = 1.0f + eps_p[0];
  const int   row16 = lane & 15;        // A-matrix row (M) this lane feeds
  const int   khalf = (lane >> 4) * 2;  // lanes 16-31 carry K+2,K+3 halves
  const int   ncol  = wave * 16 + (lane & 15);  // output column (N)
  const int   mbase = (lane >> 4) << 3; // C/D: rows M = mbase + j
  const float bias1 = b1[ncol];
  const float bias2 = b2[ncol];

  for (int blk = blockIdx.x; blk < nblocks; blk += gridDim.x) {
    const long r0 = (long)blk * 16;

    // sIn = (1+eps)*x[r0:r0+16] + acc[r0:r0+16]  (padded rows)
    {
      const float4* xv = (const float4*)(x + r0 * DIM);
      const float4* av = (const float4*)(acc + r0 * DIM);
      for (int i = tid; i < 16 * DIM / 4; i += 256) {
        int row = i >> 5;        // 32 float4 per row
        int c4  = i & 31;
        float4 xx = xv[i];
        float4 aa = av[i];
        float4 r;
        r.x = eps1 * xx.x + aa.x;
        r.y = eps1 * xx.y + aa.y;
        r.z = eps1 * xx.z + aa.z;
        r.w = eps1 * xx.w + aa.w;
        *(float4*)&sIn[row * SPAD + c4 * 4] = r;
      }
    }
    __syncthreads();

    // GEMM1: hid = relu(sIn @ W1 + b1), K accumulated 4 at a time via WMMA.
    v8f c = {0.f, 0.f, 0.f, 0.f, 0.f, 0.f, 0.f, 0.f};
#pragma unroll
    for (int k0 = 0; k0 < DIM; k0 += 4) {
      const int ka = k0 + khalf;
      v2f a;
      a.x = sIn[row16 * SPAD + ka];
      a.y = sIn[row16 * SPAD + ka + 1];
      v2f b;
      b.x = sW1[ka * HIDN + ncol];
      b.y = sW1[(ka + 1) * HIDN + ncol];
      c = __builtin_amdgcn_wmma_f32_16x16x4_f32(false, a, false, b,
                                                (short)0, c, false, false);
    }
#pragma unroll
    for (int j = 0; j < 8; ++j) {
      float hv = c[j] + bias1;
      sHid[(mbase + j) * SPAD + ncol] = hv > 0.f ? hv : 0.f;
    }
    __syncthreads();

    // GEMM2: out = sHid @ W2 + b2
    v8f c2 = {0.f, 0.f, 0.f, 0.f, 0.f, 0.f, 0.f, 0.f};
#pragma unroll
    for (int k0 = 0; k0 < HIDN; k0 += 4) {
      const int ka = k0 + khalf;
      v2f a;
      a.x = sHid[row16 * SPAD + ka];
      a.y = sHid[row16 * SPAD + ka + 1];
      v2f b;
      b.x = sW2[ka * DIM + ncol];
      b.y = sW2[(ka + 1) * DIM + ncol];
      c2 = __builtin_amdgcn_wmma_f32_16x16x4_f32(false, a, false, b,
                                                 (short)0, c2, false, false);
    }
#pragma unroll
    for (int j = 0; j < 8; ++j) {
      out[(r0 + mbase + j) * DIM + ncol] = c2[j] + bias2;
    }
    __syncthreads();
  }
}

// ---------------------------------------------------------------- launcher
extern "C" void kernel_launch(void* const* d_in, const int* in_sizes, int n_in,
                              void* d_out, int out_size, void* d_ws, size_t ws_size,
                              hipStream_t stream) {
  const float* x   = (const float*)d_in[0];
  const float* ea  = (const float*)d_in[1];
  const float* W1  = (const float*)d_in[2];
  const float* b1  = (const float*)d_in[3];
  const float* W2  = (const float*)d_in[4];
  const float* b2  = (const float*)d_in[5];
  const float* eps = (const float*)d_in[6];
  const int*   ei  = (const int*)d_in[7];
  float* out = (float*)d_out;
  float* acc = (float*)d_ws;  // N*DIM floats of scratch accumulator

  const int N = in_sizes[0] / DIM;
  const int E = in_sizes[7] / 2;

  // 1) zero the neighbor accumulator
  {
    int n4 = N * DIM / 4;
    int blocks = (n4 + 255) / 256;
    gine_zero_kernel<<<blocks, 256, 0, stream>>>((float4*)acc, n4);
  }
  // 2) edge messages + scatter-add (one wave per edge)
  {
    long threads = (long)E * 32;
    int blocks = (int)((threads + 255) / 256);
    gine_scatter_kernel<<<blocks, 256, 0, stream>>>(x, ea, ei, acc, E);
  }
  // 3) fused eps-combine + 2-layer MLP via WMMA (persistent workgroups)
  {
    int nblocks = N / 16;  // N = 100000 -> 6250 tiles of 16 nodes
    int grid = nblocks < 1024 ? nblocks : 1024;
    gine_mlp_kernel<<<grid, 256, 0, stream>>>(x, acc, W1, b1, W2, b2, eps,
                                              out, nblocks);
  }
}